// GATConv_58523224375321
// MI455X (gfx1250) — compile-verified
//
#include <hip/hip_runtime.h>

typedef float v2f __attribute__((ext_vector_type(2)));
typedef float v8f __attribute__((ext_vector_type(8)));

#define D_FEATS 64
#define NEG_SLOPE 0.2f

// Order-preserving float -> uint encoding for exact atomic max on floats.
__device__ __forceinline__ unsigned fenc(float x) {
    unsigned u = __float_as_uint(x);
    return (u & 0x80000000u) ? ~u : (u | 0x80000000u);
}
__device__ __forceinline__ float fdec(unsigned u) {
    unsigned b = (u & 0x80000000u) ? (u ^ 0x80000000u) : ~u;
    return __uint_as_float(b);
}

// ---------------- init: zero out, denom, emax ----------------
__global__ void gat_init(float* __restrict__ out, float* __restrict__ denom,
                         unsigned* __restrict__ emax, int N) {
    int i = blockIdx.x * blockDim.x + threadIdx.x;
    int total = N * D_FEATS;
    if (i < total) out[i] = 0.0f;
    if (i < N) { denom[i] = 0.0f; emax[i] = 0u; }  // 0 < fenc(-inf): safe floor
}

// ---------------- fused q/k/f projection: WMMA f32 16x16x4 ----------------
// One wave computes a 16-node x 64-out tile for all three weight matrices.
__global__ __launch_bounds__(128) void gat_proj(
    const float* __restrict__ feat,
    const float* __restrict__ Wq, const float* __restrict__ bq,
    const float* __restrict__ Wk, const float* __restrict__ bk,
    const float* __restrict__ Wf, const float* __restrict__ bf,
    float* __restrict__ q, float* __restrict__ k, float* __restrict__ f,
    int nNodes)
{
    __shared__ float Ws[3][D_FEATS][D_FEATS];   // 48 KB
    float* wsflat = &Ws[0][0][0];
    for (int i = threadIdx.x; i < D_FEATS * D_FEATS; i += 128) {
        wsflat[i]                         = Wq[i];
        wsflat[D_FEATS * D_FEATS + i]     = Wk[i];
        wsflat[2 * D_FEATS * D_FEATS + i] = Wf[i];
    }
    __syncthreads();

    int wave = threadIdx.x >> 5;
    int lane = threadIdx.x & 31;
    int tile = blockIdx.x * 4 + wave;
    if (tile * 16 >= nNodes) return;   // whole-wave uniform: EXEC stays all-1 for WMMA

    int ln = lane & 15;      // M index (A/C) or N index (B/C)
    int hi = lane >> 4;      // lane half selects K/M sub-block
    int row = tile * 16 + ln;
    if (row >= nNodes) row = nNodes - 1;   // clamp loads in tail tile
    const float* frow = feat + (size_t)row * D_FEATS;

    v8f acc[3][4];
    v8f zero = {0.f, 0.f, 0.f, 0.f, 0.f, 0.f, 0.f, 0.f};
#pragma unroll
    for (int w = 0; w < 3; ++w)
#pragma unroll
        for (int n = 0; n < 4; ++n)
            acc[w][n] = zero;

    // K loop: 16 steps of K=4. A-frag per ISA layout:
    //   lanes 0-15: K = 4t,4t+1 ; lanes 16-31: K = 4t+2,4t+3 (same rows M=0..15)
#pragma unroll
    for (int t = 0; t < 16; ++t) {
        int c0 = 4 * t + 2 * hi;
        v2f a;
        a.x = frow[c0];
        a.y = frow[c0 + 1];
#pragma unroll
        for (int w = 0; w < 3; ++w) {
#pragma unroll
            for (int n = 0; n < 4; ++n) {
                // B-frag: v0 = W[c0][N], v1 = W[c0+1][N], N = n*16 + ln
                v2f b;
                b.x = Ws[w][c0][n * 16 + ln];
                b.y = Ws[w][c0 + 1][n * 16 + ln];
                acc[w][n] = __builtin_amdgcn_wmma_f32_16x16x4_f32(
                    false, a, false, b, (short)0, acc[w][n], false, false);
            }
        }
    }

    float* outs[3] = {q, k, f};
    const float* biases[3] = {bq, bk, bf};
#pragma unroll
    for (int w = 0; w < 3; ++w) {
#pragma unroll
        for (int n = 0; n < 4; ++n) {
            int col = n * 16 + ln;
            float bias = biases[w][col];
#pragma unroll
            for (int j = 0; j < 8; ++j) {
                // C/D layout: VGPR j holds M = j + 8*hi, N = ln
                int r = tile * 16 + j + 8 * hi;
                if (r < nNodes)
                    outs[w][(size_t)r * D_FEATS + col] = acc[w][n][j] + bias;
            }
        }
    }
}

// ---------------- per-edge logits: 8 lanes per edge ----------------
__global__ __launch_bounds__(256) void gat_edge_logits(
    const float* __restrict__ q, const float* __restrict__ k,
    const int* __restrict__ src, const int* __restrict__ dst,
    float* __restrict__ ews, unsigned* __restrict__ emax, int E)
{
    int t = blockIdx.x * blockDim.x + threadIdx.x;
    int e = t >> 3;
    int g = t & 7;
    if (e >= E) return;
    int s = src[e], d = dst[e];
    const float4* kr = (const float4*)(k + (size_t)s * D_FEATS);
    const float4* qr = (const float4*)(q + (size_t)d * D_FEATS);
    float4 a0 = kr[g],     b0 = qr[g];
    float4 a1 = kr[g + 8], b1 = qr[g + 8];
    float p = a0.x * b0.x + a0.y * b0.y + a0.z * b0.z + a0.w * b0.w
            + a1.x * b1.x + a1.y * b1.y + a1.z * b1.z + a1.w * b1.w;
    p += __shfl_xor(p, 4, 8);
    p += __shfl_xor(p, 2, 8);
    p += __shfl_xor(p, 1, 8);
    if (g == 0) {
        float ev = (p > 0.0f) ? p : NEG_SLOPE * p;   // leaky relu
        ews[e] = ev;
        atomicMax(emax + d, fenc(ev));               // native integer atomic
    }
}

// ---------------- per-edge exp + denom ----------------
__global__ __launch_bounds__(256) void gat_edge_exp(
    const int* __restrict__ dst, float* __restrict__ ews,
    const unsigned* __restrict__ emax, float* __restrict__ denom, int E)
{
    int e = blockIdx.x * blockDim.x + threadIdx.x;
    if (e >= E) return;
    int d = dst[e];
    float m = fdec(emax[d]);
    float z = expf(ews[e] - m);
    ews[e] = z;
    unsafeAtomicAdd(denom + d, z);   // force native global_atomic_add_f32
}

// ---------------- weighted scatter aggregation: 64 threads per edge ----------------
__global__ __launch_bounds__(256) void gat_aggregate(
    const float* __restrict__ f, const int* __restrict__ src,
    const int* __restrict__ dst, const float* __restrict__ ews,
    const float* __restrict__ denom, float* __restrict__ out, int E)
{
    int t = blockIdx.x * blockDim.x + threadIdx.x;
    int e = t >> 6;
    int dim = t & 63;
    if (e >= E) return;
    int s = src[e], d = dst[e];
    float a = ews[e] / denom[d];
    unsafeAtomicAdd(out + (size_t)d * D_FEATS + dim,
                    a * f[(size_t)s * D_FEATS + dim]);  // native f32 atomic add
}

extern "C" void kernel_launch(void* const* d_in, const int* in_sizes, int n_in,
                              void* d_out, int out_size, void* d_ws, size_t ws_size,
                              hipStream_t stream)
{
    const float* feat = (const float*)d_in[0];
    const int*   src  = (const int*)d_in[1];
    const int*   dst  = (const int*)d_in[2];
    const float* Wq   = (const float*)d_in[3];
    const float* bq   = (const float*)d_in[4];
    const float* Wk   = (const float*)d_in[5];
    const float* bk   = (const float*)d_in[6];
    const float* Wf   = (const float*)d_in[7];
    const float* bf   = (const float*)d_in[8];
    float* out = (float*)d_out;

    int N = in_sizes[0] / D_FEATS;   // 100000
    int E = in_sizes[1];             // 1600000
    size_t N64 = (size_t)N * D_FEATS;

    // workspace layout (~84 MB)
    float*    wsf   = (float*)d_ws;
    float*    q     = wsf;
    float*    k     = wsf + N64;
    float*    f     = wsf + 2 * N64;
    float*    ews   = wsf + 3 * N64;            // E floats
    unsigned* emax  = (unsigned*)(ews + E);     // N uints
    float*    denom = (float*)(emax + N);       // N floats

    int initTotal = N * D_FEATS;
    gat_init<<<(initTotal + 255) / 256, 256, 0, stream>>>(out, denom, emax, N);

    int nTiles = (N + 15) / 16;
    gat_proj<<<(nTiles + 3) / 4, 128, 0, stream>>>(feat, Wq, bq, Wk, bk, Wf, bf,
                                                   q, k, f, N);

    long long lg = (long long)E * 8;
    gat_edge_logits<<<(unsigned)((lg + 255) / 256), 256, 0, stream>>>(
        q, k, src, dst, ews, emax, E);

    gat_edge_exp<<<(E + 255) / 256, 256, 0, stream>>>(dst, ews, emax, denom, E);

    long long ag = (long long)E * 64;
    gat_aggregate<<<(unsigned)((ag + 255) / 256), 256, 0, stream>>>(
        f, src, dst, ews, denom, out, E);
}